// PointToPixel_4346506903732
// MI455X (gfx1250) — compile-verified
//
#include <hip/hip_runtime.h>
#include <stdint.h>
#include <stddef.h>

// Problem constants (fixed by the reference implementation)
#define HH   512
#define WW   512
#define BB   8
#define NN   500000
#define HWPX (HH * WW)

#define TILE 1024   // points per block staged through LDS
#define THR  256    // threads per block (8 wave32 waves)

// ---------------------------------------------------------------------------
// CDNA5 async global->LDS copy (tracked with ASYNCcnt).
// Each call copies 16 bytes per lane from the global address in gaddr to the LDS address in lds_off.
// ---------------------------------------------------------------------------
__device__ __forceinline__ void async_g2l_b128(uint32_t lds_off, uint64_t gaddr) {
  asm volatile("global_load_async_to_lds_b128 %0, %1, off"
               :: "v"(lds_off), "v"(gaddr)
               : "memory");
}
__device__ __forceinline__ void wait_asynccnt0() {
  asm volatile("s_wait_asynccnt 0" ::: "memory");
}

// Guaranteed hardware float atomic (global_atomic_add_f32, no CAS loop).
__device__ __forceinline__ void atomAddF(float* p, float v) {
  unsafeAtomicAdd(p, v);
}

// ---------------------------------------------------------------------------
// Projection: u = round(k00*x/z + k01*y/z + k02), v = round(k10*.. + k12)
// jnp.round == round-half-to-even == rintf under default RTE mode.
// ---------------------------------------------------------------------------
__device__ __forceinline__ bool projectPoint(float px, float py, float pz,
                                             float k00, float k01, float k02,
                                             float k10, float k11, float k12,
                                             int& u, int& v) {
  float rz = 1.0f / pz;
  float xh = px * rz;
  float yh = py * rz;
  float uf = fmaf(k00, xh, fmaf(k01, yh, k02));
  float vf = fmaf(k10, xh, fmaf(k11, yh, k12));
  u = (int)rintf(uf);
  v = (int)rintf(vf);
  return (u > 0) & (u < WW) & (v > 0) & (v < HH) & (pz > 0.0f);
}

// ---------------------------------------------------------------------------
// MAIN scatter kernel (interleaved accumulator), defined FIRST and
// non-templated so the disasm snippet shows the async + atomic lowering.
// One 1024-point tile per block, staged via async->LDS:
// x tile = 12KB, c tile = 12KB (each = 256 lanes * 16B * 3 issues).
// ---------------------------------------------------------------------------
__global__ __launch_bounds__(THR) void scatter_async_inter_kernel(
    const float* __restrict__ x, const float* __restrict__ c,
    const float* __restrict__ K, float* __restrict__ acc) {
  __shared__ __align__(16) float xb[TILE * 3];
  __shared__ __align__(16) float cb[TILE * 3];

  const int       tid       = threadIdx.x;
  const long long tileStart = (long long)blockIdx.x * TILE;

  // Stage tile into LDS with async copies (ASYNCcnt path).
  {
    uint64_t gx = (uint64_t)(uintptr_t)x + (uint64_t)tileStart * 12u;
    uint64_t gc = (uint64_t)(uintptr_t)c + (uint64_t)tileStart * 12u;
    uint32_t lx = (uint32_t)(uintptr_t)xb;   // generic->LDS: low 32 bits
    uint32_t lc = (uint32_t)(uintptr_t)cb;
    uint32_t o  = (uint32_t)tid * 16u;
#pragma unroll
    for (int k = 0; k < 3; ++k) {
      uint32_t off = o + (uint32_t)k * 4096u;
      async_g2l_b128(lx + off, gx + off);
      async_g2l_b128(lc + off, gc + off);
    }
  }
  wait_asynccnt0();
  __syncthreads();

  const float k00 = K[0], k01 = K[1], k02 = K[2];
  const float k10 = K[3], k11 = K[4], k12 = K[5];

#pragma unroll
  for (int s = 0; s < TILE / THR; ++s) {
    int   i  = tid + s * THR;
    float px = xb[3 * i + 0];
    float py = xb[3 * i + 1];
    float pz = xb[3 * i + 2];
    int u, v;
    if (projectPoint(px, py, pz, k00, k01, k02, k10, k11, k12, u, v)) {
      long long gp = tileStart + i;
      int b   = (int)(gp / NN);
      int pix = v * WW + u;            // == u + v*H from the reference
      float* base = acc + ((size_t)b * HWPX + pix) * 4;
      atomAddF(base + 0, cb[3 * i + 0]);
      atomAddF(base + 1, cb[3 * i + 1]);
      atomAddF(base + 2, cb[3 * i + 2]);
      atomAddF(base + 3, 1.0f);
    }
  }
}

// Generic scatter helper for the remaining (template/tail/fallback) kernels.
template <bool INTERLEAVED>
__device__ __forceinline__ void scatterPoint(long long gp, int u, int v,
                                             float c0, float c1, float c2,
                                             float* __restrict__ img,
                                             float* __restrict__ cnt) {
  int b   = (int)(gp / NN);
  int pix = v * WW + u;
  if (INTERLEAVED) {
    float* base = img + ((size_t)b * HWPX + pix) * 4;
    atomAddF(base + 0, c0);
    atomAddF(base + 1, c1);
    atomAddF(base + 2, c2);
    atomAddF(base + 3, 1.0f);
  } else {
    size_t ib = (size_t)b * 3 * HWPX + pix;
    atomAddF(img + ib,            c0);
    atomAddF(img + ib + HWPX,     c1);
    atomAddF(img + ib + 2 * HWPX, c2);
    atomAddF(cnt + (size_t)b * HWPX + pix, 1.0f);
  }
}

// Planar-accumulator variant of the async kernel (ws-too-small fallback).
__global__ __launch_bounds__(THR) void scatter_async_planar_kernel(
    const float* __restrict__ x, const float* __restrict__ c,
    const float* __restrict__ K,
    float* __restrict__ img, float* __restrict__ cnt) {
  __shared__ __align__(16) float xb[TILE * 3];
  __shared__ __align__(16) float cb[TILE * 3];

  const int       tid       = threadIdx.x;
  const long long tileStart = (long long)blockIdx.x * TILE;
  {
    uint64_t gx = (uint64_t)(uintptr_t)x + (uint64_t)tileStart * 12u;
    uint64_t gc = (uint64_t)(uintptr_t)c + (uint64_t)tileStart * 12u;
    uint32_t lx = (uint32_t)(uintptr_t)xb;
    uint32_t lc = (uint32_t)(uintptr_t)cb;
    uint32_t o  = (uint32_t)tid * 16u;
#pragma unroll
    for (int k = 0; k < 3; ++k) {
      uint32_t off = o + (uint32_t)k * 4096u;
      async_g2l_b128(lx + off, gx + off);
      async_g2l_b128(lc + off, gc + off);
    }
  }
  wait_asynccnt0();
  __syncthreads();

  const float k00 = K[0], k01 = K[1], k02 = K[2];
  const float k10 = K[3], k11 = K[4], k12 = K[5];

#pragma unroll
  for (int s = 0; s < TILE / THR; ++s) {
    int   i  = tid + s * THR;
    float px = xb[3 * i + 0];
    float py = xb[3 * i + 1];
    float pz = xb[3 * i + 2];
    int u, v;
    if (projectPoint(px, py, pz, k00, k01, k02, k10, k11, k12, u, v))
      scatterPoint<false>(tileStart + i, u, v,
                          cb[3 * i + 0], cb[3 * i + 1], cb[3 * i + 2], img, cnt);
  }
}

// ---------------------------------------------------------------------------
// Tail kernel: direct global loads for the (< TILE) leftover points, so the
// async b128 staging never reads past the end of the input buffers.
// ---------------------------------------------------------------------------
template <bool INTERLEAVED>
__global__ __launch_bounds__(THR) void scatter_direct_kernel(
    const float* __restrict__ x, const float* __restrict__ c,
    const float* __restrict__ K,
    float* __restrict__ img, float* __restrict__ cnt,
    long long start, long long count) {
  long long idx = (long long)blockIdx.x * blockDim.x + threadIdx.x;
  if (idx >= count) return;
  long long gp = start + idx;

  const float k00 = K[0], k01 = K[1], k02 = K[2];
  const float k10 = K[3], k11 = K[4], k12 = K[5];

  float px = x[3 * gp + 0];
  float py = x[3 * gp + 1];
  float pz = x[3 * gp + 2];
  int u, v;
  if (projectPoint(px, py, pz, k00, k01, k02, k10, k11, k12, u, v)) {
    scatterPoint<INTERLEAVED>(gp, u, v,
                              c[3 * gp + 0], c[3 * gp + 1], c[3 * gp + 2],
                              img, cnt);
  }
}

// ---------------------------------------------------------------------------
// Normalize: out[b,ch,pix] = acc[b,pix,ch] / max(acc[b,pix,3], 1)
// One exact division + 3 multiplies. Fully overwrites d_out.
// ---------------------------------------------------------------------------
__global__ __launch_bounds__(THR) void normalize_interleaved_kernel(
    const float4* __restrict__ ws, float* __restrict__ out) {
  unsigned i = blockIdx.x * blockDim.x + threadIdx.x;
  if (i >= (unsigned)(BB * HWPX)) return;
  float4 w = ws[i];
  float  inv = 1.0f / fmaxf(w.w, 1.0f);
  unsigned b = i / HWPX;
  unsigned p = i % HWPX;
  size_t ob = (size_t)b * 3 * HWPX + p;
  out[ob]            = w.x * inv;
  out[ob + HWPX]     = w.y * inv;
  out[ob + 2 * HWPX] = w.z * inv;
}

__global__ __launch_bounds__(THR) void normalize_planar_kernel(
    float* __restrict__ out, const float* __restrict__ cnt) {
  unsigned i = blockIdx.x * blockDim.x + threadIdx.x;
  if (i >= (unsigned)(BB * HWPX)) return;
  float inv = 1.0f / fmaxf(cnt[i], 1.0f);
  unsigned b = i / HWPX;
  unsigned p = i % HWPX;
  size_t ob = (size_t)b * 3 * HWPX + p;
  out[ob]            = out[ob] * inv;
  out[ob + HWPX]     = out[ob + HWPX] * inv;
  out[ob + 2 * HWPX] = out[ob + 2 * HWPX] * inv;
}

// ---------------------------------------------------------------------------
extern "C" void kernel_launch(void* const* d_in, const int* in_sizes, int n_in,
                              void* d_out, int out_size, void* d_ws, size_t ws_size,
                              hipStream_t stream) {
  const float* x = (const float*)d_in[0];   // (B, N, 3) f32
  const float* c = (const float*)d_in[1];   // (B, N, 3) f32
  const float* K = (const float*)d_in[2];   // (3, 3)    f32
  float* out = (float*)d_out;               // (B, 3, H, W) f32

  const long long P         = (long long)in_sizes[0] / 3;   // B*N points
  const long long fullTiles = P / TILE;
  const long long tail      = P - fullTiles * TILE;

  const size_t needInterleaved = (size_t)BB * HWPX * 4 * sizeof(float);  // 32 MB
  const size_t needCountOnly   = (size_t)BB * HWPX * sizeof(float);      // 8 MB

  if (ws_size >= needInterleaved) {
    float* acc = (float*)d_ws;  // (B, H*W, 4): c0,c1,c2,count
    hipMemsetAsync(acc, 0, needInterleaved, stream);
    if (fullTiles > 0)
      scatter_async_inter_kernel<<<(unsigned)fullTiles, THR, 0, stream>>>(x, c, K, acc);
    if (tail > 0)
      scatter_direct_kernel<true><<<(unsigned)((tail + THR - 1) / THR), THR, 0, stream>>>(
          x, c, K, acc, nullptr, fullTiles * TILE, tail);
    normalize_interleaved_kernel<<<(BB * HWPX + THR - 1) / THR, THR, 0, stream>>>(
        (const float4*)acc, out);
  } else {
    // Fallback: accumulate colors planar directly in d_out, counts in d_ws.
    float* cnt = (float*)d_ws;  // (B, H*W)
    hipMemsetAsync(out, 0, (size_t)BB * 3 * HWPX * sizeof(float), stream);
    hipMemsetAsync(cnt, 0, needCountOnly, stream);
    if (fullTiles > 0)
      scatter_async_planar_kernel<<<(unsigned)fullTiles, THR, 0, stream>>>(x, c, K, out, cnt);
    if (tail > 0)
      scatter_direct_kernel<false><<<(unsigned)((tail + THR - 1) / THR), THR, 0, stream>>>(
          x, c, K, out, cnt, fullTiles * TILE, tail);
    normalize_planar_kernel<<<(BB * HWPX + THR - 1) / THR, THR, 0, stream>>>(out, cnt);
  }
}